// PersistentHomology_81853486727576
// MI455X (gfx1250) — compile-verified
//
#include <hip/hip_runtime.h>
#include <math.h>

typedef __attribute__((ext_vector_type(2))) float v2f;
typedef __attribute__((ext_vector_type(8))) float v8f;

#define D_DIM 128   // embedding dim (reference: 128)

// ---------------------------------------------------------------------------
// Kernel 1: row squared-norms + zero the global double accumulators
// ---------------------------------------------------------------------------
__global__ void ph_sqnorm_init(const float* __restrict__ X,
                               float* __restrict__ sq,
                               double* __restrict__ S,
                               int nRows) {
    if (blockIdx.x == 0 && threadIdx.x < 2) S[threadIdx.x] = 0.0;
    int r = blockIdx.x * blockDim.x + threadIdx.x;
    if (r >= nRows) return;
    const float4* p = (const float4*)(X + (size_t)r * D_DIM);
    float s = 0.f;
#pragma unroll
    for (int i = 0; i < D_DIM / 4; ++i) {
        float4 v = p[i];
        s += v.x * v.x + v.y * v.y + v.z * v.z + v.w * v.w;
    }
    sq[r] = s;
}

// ---------------------------------------------------------------------------
// Kernel 2: fused Gram (FP32 WMMA) -> distances -> (sum, sum_sq) reduction.
// Block = 256 threads = 8 waves. Wave handles 2 adjacent 16x16 j-tiles for
// one i-tile. A and B strips are fully preloaded into registers so each
// 32-deep WMMA chain issues back-to-back behind a single s_wait_loadcnt.
// Only j-tile >= i-tile computed; strictly-upper tiles weighted x2.
// ---------------------------------------------------------------------------
__global__ __launch_bounds__(256) void ph_gram_kernel(
    const float* __restrict__ X,
    const float* __restrict__ sq,
    double* __restrict__ S,
    int nRows)
{
    const int it = (int)blockIdx.y;                    // i tile index
    // whole block strictly below the tile-diagonal -> nothing to do (uniform)
    if ((int)blockIdx.x * 16 + 15 < it) return;

    const int lane  = threadIdx.x & 31;
    const int wid   = threadIdx.x >> 5;
    const int mlane = lane & 15;                       // M (for A) / N (for B/C)
    const int khalf = lane >> 4;                       // selects K pair {0,1} vs {2,3}

    const int i0 = it * 16;

    // Preload A strip: A[M=mlane][K = 4*kk + 2*khalf + v], v=0..1, kk=0..31
    const float* rowA = X + (size_t)(i0 + mlane) * D_DIM + 2 * khalf;
    v2f a[32];
#pragma unroll
    for (int kk = 0; kk < 32; ++kk)
        a[kk] = *(const v2f*)(rowA + 4 * kk);

    float s1 = 0.f, s2 = 0.f;
    const int jtBase = ((int)blockIdx.x * 8 + wid) * 2;

#pragma unroll
    for (int jj = 0; jj < 2; ++jj) {
        const int jt = jtBase + jj;
        if (jt < it) continue;                         // wave-uniform skip
        const int j0 = jt * 16;

        // Preload full B strip for this j-tile (same per-lane layout as A)
        const float* rowB = X + (size_t)(j0 + mlane) * D_DIM + 2 * khalf;
        v2f b[32];
#pragma unroll
        for (int kk = 0; kk < 32; ++kk)
            b[kk] = *(const v2f*)(rowB + 4 * kk);

        // Uninterrupted WMMA chain on two independent accumulators
        v8f acc0 = {}; v8f acc1 = {};
#pragma unroll
        for (int kk = 0; kk < 32; kk += 2) {
            acc0 = __builtin_amdgcn_wmma_f32_16x16x4_f32(
                false, a[kk],     false, b[kk],     (short)0, acc0, false, false);
            acc1 = __builtin_amdgcn_wmma_f32_16x16x4_f32(
                false, a[kk + 1], false, b[kk + 1], (short)0, acc1, false, false);
        }

        // C/D layout: component r, lane -> (M = r + 8*khalf, N = mlane)
        const float w  = (jt == it) ? 1.f : 2.f;
        const float sn = sq[j0 + mlane];
        float ls1 = 0.f, ls2 = 0.f;
#pragma unroll
        for (int r = 0; r < 8; ++r) {
            const int gi = i0 + r + 8 * khalf;
            float dot = acc0[r] + acc1[r];
            float d2  = sq[gi] + sn - 2.f * dot;
            d2 = fmaxf(d2, 0.f);                       // safe sqrt: exact 0 on diag
            ls1 += sqrtf(d2);
            ls2 += d2;                                  // d^2 feeds the variance term
        }
        s1 += w * ls1;
        s2 += w * ls2;
    }

    // wave32 shuffle reduction
#pragma unroll
    for (int off = 16; off > 0; off >>= 1) {
        s1 += __shfl_down(s1, off, 32);
        s2 += __shfl_down(s2, off, 32);
    }

    __shared__ double red1[8];
    __shared__ double red2[8];
    if (lane == 0) { red1[wid] = (double)s1; red2[wid] = (double)s2; }
    __syncthreads();
    if (threadIdx.x == 0) {
        double t1 = 0.0, t2 = 0.0;
#pragma unroll
        for (int i = 0; i < 8; ++i) { t1 += red1[i]; t2 += red2[i]; }
        atomicAdd(&S[0], t1);    // global_atomic_add_f64
        atomicAdd(&S[1], t2);
    }
}

// ---------------------------------------------------------------------------
// Kernel 3: finalize scalar  std(ddof=1) / (mean + 1e-8)
// ---------------------------------------------------------------------------
__global__ void ph_finalize(const double* __restrict__ S,
                            float* __restrict__ out,
                            int nRows) {
    if (threadIdx.x != 0 || blockIdx.x != 0) return;
    const double Ntot = (double)nRows * (double)nRows;
    const double s1 = S[0], s2 = S[1];
    const double mean = s1 / Ntot;
    double var = (s2 - s1 * s1 / Ntot) / (Ntot - 1.0);
    var = var > 0.0 ? var : 0.0;
    out[0] = (float)(sqrt(var) / (mean + 1e-8));
}

// ---------------------------------------------------------------------------
extern "C" void kernel_launch(void* const* d_in, const int* in_sizes, int n_in,
                              void* d_out, int out_size, void* d_ws, size_t ws_size,
                              hipStream_t stream) {
    const float* X = (const float*)d_in[0];
    float* out = (float*)d_out;
    const int nRows = in_sizes[0] / D_DIM;            // 4*2048 = 8192

    double* S  = (double*)d_ws;                       // [0]=sum d, [1]=sum d^2
    float*  sq = (float*)((char*)d_ws + 64);          // row squared norms

    ph_sqnorm_init<<<(nRows + 255) / 256, 256, 0, stream>>>(X, sq, S, nRows);

    const int T = nRows / 16;                         // 512 tiles per dim
    dim3 grid(T / 16, T);                             // (32, 512)
    ph_gram_kernel<<<grid, 256, 0, stream>>>(X, sq, S, nRows);

    ph_finalize<<<1, 1, 0, stream>>>(S, out, nRows);
}